// SoftAgg_8873402434226
// MI455X (gfx1250) — compile-verified
//
#include <hip/hip_runtime.h>

typedef float v2f __attribute__((ext_vector_type(2)));
typedef float v8f __attribute__((ext_vector_type(8)));
typedef int   v4i __attribute__((vector_size(16)));

#if defined(__has_builtin)
#if __has_builtin(__builtin_amdgcn_global_load_async_to_lds_b128) && \
    __has_builtin(__builtin_amdgcn_s_wait_asynccnt)
#define USE_ASYNC 1
#endif
#endif
#ifndef USE_ASYNC
#define USE_ASYNC 0
#endif

#if USE_ASYNC
typedef __attribute__((address_space(1))) v4i gv4i;  // global int4
typedef __attribute__((address_space(3))) v4i lv4i;  // LDS int4
#define ASYNC_B128(gp, lp) \
  __builtin_amdgcn_global_load_async_to_lds_b128((gv4i*)(gp), (lv4i*)(lp), 0, 0)
#endif

namespace {
constexpr int Bn = 8;
constexpr int Nn = 65536;
constexpr int Dn = 512;
constexpr int Sn = 512;          // NUM_SEG upper bound; empty segments never gathered
constexpr int M1 = Bn * Nn;      // 524288 rows for the big GEMMs
constexpr int M2 = Sn * Bn;      // 4096 rows for the hy GEMM

// Order-preserving float <-> uint mapping so atomicMax(uint) == float max (exact).
__device__ __forceinline__ unsigned f2ord(float f) {
  unsigned u = __float_as_uint(f);
  return (u & 0x80000000u) ? ~u : (u | 0x80000000u);
}
__device__ __forceinline__ float ord2f(unsigned u) {
  return __uint_as_float((u & 0x80000000u) ? (u ^ 0x80000000u) : ~u);
}
constexpr unsigned ORD_NEG_INF = 0x007FFFFFu;  // f2ord(-inf)
}  // namespace

// ---------------------------------------------------------------------------
// GEMM 1: Gx = X*Wg^T + bg, Fx = X*Wf^T + bf   (X tile in LDS reused for both)
// Block tile 64(M) x 64(E); 8 waves; wave = 16(M) x 32(E). K staged 32 at a
// time via async global->LDS copies, double-buffered: chunk i+1 copies overlap
// chunk i's 32 v_wmma_f32_16x16x4_f32 per wave.
// ---------------------------------------------------------------------------
__global__ __launch_bounds__(256, 2) void k_gemm_gf(
    const float* __restrict__ X,
    const float* __restrict__ Wg, const float* __restrict__ bgv,
    const float* __restrict__ Wf, const float* __restrict__ bfv,
    float* __restrict__ Gx, float* __restrict__ Fx)
{
  __shared__ __align__(16) float Xs[2][64][36];
  __shared__ __align__(16) float Gs[2][64][36];
  __shared__ __align__(16) float Fs[2][64][36];

  const int tid  = threadIdx.x;
  const int lane = tid & 31;
  const int wave = tid >> 5;
  const int wm   = (wave & 3) * 16;     // wave's M subtile within block tile
  const int we   = (wave >> 2) * 32;    // wave's E half within block tile
  const int l15  = lane & 15;
  const int hsel = (lane >> 4) * 2;     // K sub-offset (0 or 2) per A/B layout
  const size_t row0 = (size_t)blockIdx.x * 64;
  const int    e0   = blockIdx.y * 64;

  v8f aG0 = {}, aG1 = {}, aF0 = {}, aF1 = {};

  auto compute = [&](int buf) {
#pragma unroll
    for (int k8 = 0; k8 < 8; ++k8) {
      const int kk = k8 * 4 + hsel;
      const v2f a   = *(const v2f*)&Xs[buf][wm + l15][kk];
      const v2f bg0 = *(const v2f*)&Gs[buf][we + l15][kk];
      const v2f bg1 = *(const v2f*)&Gs[buf][we + 16 + l15][kk];
      const v2f bf0 = *(const v2f*)&Fs[buf][we + l15][kk];
      const v2f bf1 = *(const v2f*)&Fs[buf][we + 16 + l15][kk];
      aG0 = __builtin_amdgcn_wmma_f32_16x16x4_f32(false, a, false, bg0, (short)0, aG0, false, false);
      aG1 = __builtin_amdgcn_wmma_f32_16x16x4_f32(false, a, false, bg1, (short)0, aG1, false, false);
      aF0 = __builtin_amdgcn_wmma_f32_16x16x4_f32(false, a, false, bf0, (short)0, aF0, false, false);
      aF1 = __builtin_amdgcn_wmma_f32_16x16x4_f32(false, a, false, bf1, (short)0, aF1, false, false);
    }
  };

#if USE_ASYNC
  auto issue = [&](int buf, int kc) {
#pragma unroll
    for (int i = 0; i < 2; ++i) {
      const int q = tid * 2 + i;        // 512 b128 slots per tile
      const int r = q >> 3;
      const int c = (q & 7) * 4;
      ASYNC_B128(X  + (row0 + r) * Dn + kc + c,          &Xs[buf][r][c]);
      ASYNC_B128(Wg + (size_t)(e0 + r) * Dn + kc + c,    &Gs[buf][r][c]);
      ASYNC_B128(Wf + (size_t)(e0 + r) * Dn + kc + c,    &Fs[buf][r][c]);
    }
  };
  issue(0, 0);
#pragma unroll 1
  for (int it = 0; it < 16; ++it) {
    __builtin_amdgcn_s_wait_asynccnt(0);   // chunk it landed in LDS
    __syncthreads();                        // ..and everyone is done with 1-buf
    if (it < 15) issue((it + 1) & 1, (it + 1) * 32);  // overlaps with compute
    compute(it & 1);
  }
#else
  for (int it = 0; it < 16; ++it) {
    const int kc = it * 32;
#pragma unroll
    for (int i = 0; i < 2; ++i) {
      const int q = tid * 2 + i;
      const int r = q >> 3;
      const int c = (q & 7) * 4;
      *(float4*)&Xs[0][r][c] = *(const float4*)(X  + (row0 + r) * Dn + kc + c);
      *(float4*)&Gs[0][r][c] = *(const float4*)(Wg + (size_t)(e0 + r) * Dn + kc + c);
      *(float4*)&Fs[0][r][c] = *(const float4*)(Wf + (size_t)(e0 + r) * Dn + kc + c);
    }
    __syncthreads();
    compute(0);
    __syncthreads();
  }
#endif

  const int c0 = e0 + we + l15;
  const int c1 = c0 + 16;
  const float bG0 = bgv[c0], bG1 = bgv[c1], bF0 = bfv[c0], bF1 = bfv[c1];
  const size_t rb = row0 + wm + (size_t)(lane >> 4) * 8;  // C layout: M = i + 8*half
#pragma unroll
  for (int i = 0; i < 8; ++i) {
    const size_t ro = (rb + i) * Dn;
    Gx[ro + c0] = aG0[i] + bG0;
    Gx[ro + c1] = aG1[i] + bG1;
    Fx[ro + c0] = aF0[i] + bF0;
    Fx[ro + c1] = aF1[i] + bF1;
  }
}

// ---------------------------------------------------------------------------
// GEMM 2: Hy = Y*Wh^T + bh over [S*B, D]  (Y precomputed by k_seg_y)
// ---------------------------------------------------------------------------
__global__ __launch_bounds__(256, 2) void k_gemm_h(
    const float* __restrict__ Y, const float* __restrict__ Wh,
    const float* __restrict__ bhv, float* __restrict__ Hy)
{
  __shared__ __align__(16) float Ys[2][64][36];
  __shared__ __align__(16) float Hs[2][64][36];

  const int tid  = threadIdx.x;
  const int lane = tid & 31;
  const int wave = tid >> 5;
  const int wm   = (wave & 3) * 16;
  const int we   = (wave >> 2) * 32;
  const int l15  = lane & 15;
  const int hsel = (lane >> 4) * 2;
  const size_t row0 = (size_t)blockIdx.x * 64;
  const int    e0   = blockIdx.y * 64;

  v8f aH0 = {}, aH1 = {};

  auto compute = [&](int buf) {
#pragma unroll
    for (int k8 = 0; k8 < 8; ++k8) {
      const int kk = k8 * 4 + hsel;
      const v2f a  = *(const v2f*)&Ys[buf][wm + l15][kk];
      const v2f b0 = *(const v2f*)&Hs[buf][we + l15][kk];
      const v2f b1 = *(const v2f*)&Hs[buf][we + 16 + l15][kk];
      aH0 = __builtin_amdgcn_wmma_f32_16x16x4_f32(false, a, false, b0, (short)0, aH0, false, false);
      aH1 = __builtin_amdgcn_wmma_f32_16x16x4_f32(false, a, false, b1, (short)0, aH1, false, false);
    }
  };

#if USE_ASYNC
  auto issue = [&](int buf, int kc) {
#pragma unroll
    for (int i = 0; i < 2; ++i) {
      const int q = tid * 2 + i;
      const int r = q >> 3;
      const int c = (q & 7) * 4;
      ASYNC_B128(Y  + (row0 + r) * Dn + kc + c,       &Ys[buf][r][c]);
      ASYNC_B128(Wh + (size_t)(e0 + r) * Dn + kc + c, &Hs[buf][r][c]);
    }
  };
  issue(0, 0);
#pragma unroll 1
  for (int it = 0; it < 16; ++it) {
    __builtin_amdgcn_s_wait_asynccnt(0);
    __syncthreads();
    if (it < 15) issue((it + 1) & 1, (it + 1) * 32);
    compute(it & 1);
  }
#else
  for (int it = 0; it < 16; ++it) {
    const int kc = it * 32;
#pragma unroll
    for (int i = 0; i < 2; ++i) {
      const int q = tid * 2 + i;
      const int r = q >> 3;
      const int c = (q & 7) * 4;
      *(float4*)&Ys[0][r][c] = *(const float4*)(Y  + (row0 + r) * Dn + kc + c);
      *(float4*)&Hs[0][r][c] = *(const float4*)(Wh + (size_t)(e0 + r) * Dn + kc + c);
    }
    __syncthreads();
    compute(0);
    __syncthreads();
  }
#endif

  const int c0 = e0 + we + l15;
  const int c1 = c0 + 16;
  const float b0 = bhv[c0], b1 = bhv[c1];
  const size_t rb = row0 + wm + (size_t)(lane >> 4) * 8;
#pragma unroll
  for (int i = 0; i < 8; ++i) {
    const size_t ro = (rb + i) * Dn;
    Hy[ro + c0] = aH0[i] + b0;
    Hy[ro + c1] = aH1[i] + b1;
  }
}

// ---------------------------------------------------------------------------
// Segment-stat kernels (stat buffers are 8.4 MB each -> L2-resident atomics)
// ---------------------------------------------------------------------------
__global__ void k_seg_init(unsigned* __restrict__ Mf, float* __restrict__ Den,
                           float* __restrict__ Yn) {
  const int i = blockIdx.x * blockDim.x + threadIdx.x;
  Mf[i] = ORD_NEG_INF;
  Den[i] = 0.f;
  Yn[i] = 0.f;
}

__global__ void k_seg_max(const float* __restrict__ Gx, const int* __restrict__ ix,
                          unsigned* __restrict__ Mf) {
  const size_t t = (size_t)blockIdx.x * blockDim.x + threadIdx.x;  // (b*N+n)*(D/4)+e4
  const int e4 = (int)(t & (Dn / 4 - 1));
  const size_t rowbn = t >> 7;            // D/4 == 128
  const int n = (int)(rowbn & (Nn - 1));
  const int b = (int)(rowbn >> 16);       // N == 65536
  const int s = ix[n];
  const float4 g = *(const float4*)(Gx + rowbn * Dn + (size_t)e4 * 4);
  unsigned* dst = Mf + ((size_t)(s * Bn + b) * Dn) + (size_t)e4 * 4;
  atomicMax(dst + 0, f2ord(g.x));
  atomicMax(dst + 1, f2ord(g.y));
  atomicMax(dst + 2, f2ord(g.z));
  atomicMax(dst + 3, f2ord(g.w));
}

__global__ void k_seg_acc(const float* __restrict__ Gx, const float* __restrict__ Fx,
                          const int* __restrict__ ix, const unsigned* __restrict__ Mf,
                          float* __restrict__ Den, float* __restrict__ Yn) {
  const size_t t = (size_t)blockIdx.x * blockDim.x + threadIdx.x;
  const int e4 = (int)(t & (Dn / 4 - 1));
  const size_t rowbn = t >> 7;
  const int n = (int)(rowbn & (Nn - 1));
  const int b = (int)(rowbn >> 16);
  const int s = ix[n];
  const size_t go = rowbn * Dn + (size_t)e4 * 4;
  const float4 g = *(const float4*)(Gx + go);
  const float4 f = *(const float4*)(Fx + go);
  const size_t so = ((size_t)(s * Bn + b) * Dn) + (size_t)e4 * 4;
  float ex[4];
  ex[0] = expf(g.x - ord2f(Mf[so + 0]));
  ex[1] = expf(g.y - ord2f(Mf[so + 1]));
  ex[2] = expf(g.z - ord2f(Mf[so + 2]));
  ex[3] = expf(g.w - ord2f(Mf[so + 3]));
  const float fv[4] = {f.x, f.y, f.z, f.w};
#pragma unroll
  for (int j = 0; j < 4; ++j) {
    __hip_atomic_fetch_add(&Den[so + j], ex[j], __ATOMIC_RELAXED, __HIP_MEMORY_SCOPE_AGENT);
    __hip_atomic_fetch_add(&Yn[so + j], ex[j] * fv[j], __ATOMIC_RELAXED, __HIP_MEMORY_SCOPE_AGENT);
  }
}

// y = ynum/denom (0 for empty segments), 2M elems, L2-resident
__global__ void k_seg_y(const float* __restrict__ Yn, const float* __restrict__ Den,
                        float* __restrict__ Y) {
  const int i = blockIdx.x * blockDim.x + threadIdx.x;
  const float d = Den[i];
  Y[i] = d > 0.f ? Yn[i] / d : 0.f;
}

__global__ void k_gather(const float* __restrict__ Hy, const int* __restrict__ ix,
                         float* __restrict__ out) {
  const size_t t = (size_t)blockIdx.x * blockDim.x + threadIdx.x;
  const int e4 = (int)(t & (Dn / 4 - 1));
  const size_t rowbn = t >> 7;
  const int n = (int)(rowbn & (Nn - 1));
  const int b = (int)(rowbn >> 16);
  const int s = ix[n];
  const float4 v = *(const float4*)(Hy + ((size_t)(s * Bn + b) * Dn) + (size_t)e4 * 4);
  *(float4*)(out + rowbn * Dn + (size_t)e4 * 4) = v;
}

// ---------------------------------------------------------------------------
extern "C" void kernel_launch(void* const* d_in, const int* in_sizes, int n_in,
                              void* d_out, int out_size, void* d_ws, size_t ws_size,
                              hipStream_t stream) {
  const float* x  = (const float*)d_in[0];
  const int*   ix = (const int*)d_in[1];   // integer input -> const int*
  const float* Wf = (const float*)d_in[2];
  const float* bf = (const float*)d_in[3];
  const float* Wg = (const float*)d_in[4];
  const float* bg = (const float*)d_in[5];
  const float* Wh = (const float*)d_in[6];
  const float* bh = (const float*)d_in[7];
  float* out = (float*)d_out;

  char* w = (char*)d_ws;
  float*    Gx  = (float*)w;    w += (size_t)M1 * Dn * sizeof(float);      // 1.07 GB
  float*    Fx  = (float*)w;    w += (size_t)M1 * Dn * sizeof(float);      // 1.07 GB
  unsigned* Mf  = (unsigned*)w; w += (size_t)Sn * Bn * Dn * sizeof(unsigned);
  float*    Den = (float*)w;    w += (size_t)Sn * Bn * Dn * sizeof(float);
  float*    Yn  = (float*)w;    w += (size_t)Sn * Bn * Dn * sizeof(float);
  float*    Yv  = (float*)w;    w += (size_t)Sn * Bn * Dn * sizeof(float);
  float*    Hy  = (float*)w;

  const int sbd = Sn * Bn * Dn;                    // 2,097,152
  const size_t tq = (size_t)M1 * (Dn / 4);         // 67,108,864 (div by 256)

  k_gemm_gf<<<dim3(M1 / 64, Dn / 64), 256, 0, stream>>>(x, Wg, bg, Wf, bf, Gx, Fx);
  k_seg_init<<<sbd / 256, 256, 0, stream>>>(Mf, Den, Yn);
  k_seg_max<<<(unsigned)(tq / 256), 256, 0, stream>>>(Gx, ix, Mf);
  k_seg_acc<<<(unsigned)(tq / 256), 256, 0, stream>>>(Gx, Fx, ix, Mf, Den, Yn);
  k_seg_y<<<sbd / 256, 256, 0, stream>>>(Yn, Den, Yv);
  k_gemm_h<<<dim3(M2 / 64, Dn / 64), 256, 0, stream>>>(Yv, Wh, bh, Hy);
  k_gather<<<(unsigned)(tq / 256), 256, 0, stream>>>(Hy, ix, out);
}